// DecodePredictions_59931973648595
// MI455X (gfx1250) — compile-verified
//
#include <hip/hip_runtime.h>
#include <stdint.h>
#include <math.h>

// ---------------- problem constants (match reference) ----------------
#define NA        76725      // total anchors for 640x640 (levels 3..7)
#define NC        80         // classes
#define NB        8          // batch
#define PRED_W    84         // 4 box + 80 class logits
#define TOPK      100
#define NT        256        // threads per block (8 wave32)
#define NBINS     2048
#define CAP       2048       // candidate capacity (pow2 for bitonic)
#define CONF_THRF 0.05f
#define IOU_THRF  0.5f

// ---------------- helpers ----------------
__device__ __forceinline__ float sigmoidf_(float x) {
    return 1.0f / (1.0f + __expf(-x));
}

__device__ __forceinline__ int logit_bin(float x) {
    x = fminf(fmaxf(x, -12.0f), 12.0f);
    int b = (int)((x + 12.0f) * ((float)NBINS / 24.0f));
    return b < 0 ? 0 : (b > NBINS - 1 ? NBINS - 1 : b);
}

__device__ __forceinline__ int score_bin(float v) {
    int b = (int)(v * (float)NBINS);
    return b < 0 ? 0 : (b > NBINS - 1 ? NBINS - 1 : b);
}

// Anchor (cx, cy, w, h) from flat anchor index; faithful to AnchorBox.get_anchors.
__device__ __forceinline__ void anchor_at(int a, float& cx, float& cy,
                                          float& aw, float& ah) {
    int fw, rem; float stride, area;
    if (a < 57600)      { rem = a;         fw = 80; stride =   8.f; area =   1024.f; }
    else if (a < 72000) { rem = a - 57600; fw = 40; stride =  16.f; area =   4096.f; }
    else if (a < 75600) { rem = a - 72000; fw = 20; stride =  32.f; area =  16384.f; }
    else if (a < 76500) { rem = a - 75600; fw = 10; stride =  64.f; area =  65536.f; }
    else                { rem = a - 76500; fw =  5; stride = 128.f; area = 262144.f; }
    int d    = rem % 9;
    int cell = rem / 9;
    int x = cell % fw;
    int y = cell / fw;
    cx = ((float)x + 0.5f) * stride;
    cy = ((float)y + 0.5f) * stride;
    float r = (d < 3) ? 0.5f : (d < 6 ? 1.0f : 2.0f);
    int   si = d % 3;
    float s  = (si == 0) ? 1.0f : (si == 1 ? 1.2599210499f : 1.5874010520f);
    float h = __fsqrt_rn(area / r);
    float w = area / h;
    aw = s * w;
    ah = s * h;
}

// All threads compute (identically) the largest bin t such that
// sum(hist[t..NBINS-1]) >= want. hist must be complete (barrier by caller).
__device__ int find_thr_bin(const int* hist, int* strip, int tid, int want) {
    const int SPT = NBINS / NT;  // 8 bins per thread
    int ssum = 0;
#pragma unroll
    for (int i = 0; i < SPT; ++i) ssum += hist[tid * SPT + i];
    strip[tid] = ssum;
    __syncthreads();
    int acc = 0, sIdx = -1;
    for (int t = NT - 1; t >= 0; --t) {
        acc += strip[t];
        if (acc >= want) { sIdx = t; break; }
    }
    if (sIdx < 0) return 0;                 // fewer than `want` entries total
    int accAbove = acc - strip[sIdx];
    int thr = sIdx * SPT;
    for (int t = (sIdx + 1) * SPT - 1; t >= sIdx * SPT; --t) {
        accAbove += hist[t];
        if (accAbove >= want) { thr = t; break; }
    }
    return thr;
}

// In-LDS bitonic sort, descending by key, CAP elements, NT threads.
__device__ void bitonic_desc(float* key, int* val, int tid) {
    for (int k = 2; k <= CAP; k <<= 1) {
        for (int j = k >> 1; j > 0; j >>= 1) {
            for (int i = tid; i < CAP; i += NT) {
                int ixj = i ^ j;
                if (ixj > i) {
                    bool desc = ((i & k) == 0);
                    float a = key[i], b = key[ixj];
                    bool sw = desc ? (a < b) : (a > b);
                    if (sw) {
                        key[i] = b; key[ixj] = a;
                        int t = val[i]; val[i] = val[ixj]; val[ixj] = t;
                    }
                }
            }
            __syncthreads();
        }
    }
}

// ---------------- Kernel A: per-(image,class) top-100 + NMS ----------------
__global__ void __launch_bounds__(NT)
topk_nms_per_class(const float* __restrict__ preds,
                   float* __restrict__ wsScores,   // [NB*NC*TOPK]
                   float* __restrict__ wsBoxes) {  // [NB*NC*TOPK*4]
    const int c   = blockIdx.x;
    const int b   = blockIdx.y;
    const int tid = threadIdx.x;

    __shared__ int          hist[NBINS];
    __shared__ int          strip[NT];
    __shared__ float        candL[CAP];
    __shared__ int          candI[CAP];
    __shared__ float        stage[2][NT];   // double-buffered async staging
    __shared__ int          cnt;
    __shared__ float        sc[TOPK];
    __shared__ float        bxs[4][TOPK];
    __shared__ unsigned int iouMask[TOPK][4];
    __shared__ unsigned int keepW[4];

    for (int i = tid; i < NBINS; i += NT) hist[i] = 0;
    if (tid == 0) cnt = 0;
    __syncthreads();

    // Logit column base for (b, c): stride PRED_W floats between anchors.
    const float* basep = preds + (size_t)b * NA * PRED_W + 4 + c;

    // ---- Pass 1: histogram of logits, streamed with the gfx1250 async
    // global->LDS pipeline (ASYNCcnt), software-pipelined (double buffer):
    // issue tile t+1, wait asynccnt<=1 (in-order retirement => tile t done),
    // consume tile t. Final tile waits on 0. ----
    const int numTiles = (NA + NT - 1) / NT;
    {
        unsigned int voff = (unsigned int)tid * (unsigned int)(PRED_W * 4);
        unsigned int lds0 = (unsigned int)(uintptr_t)&stage[0][tid];
        asm volatile("global_load_async_to_lds_b32 %0, %1, %2"
                     :: "v"(lds0), "v"(voff), "s"(basep) : "memory");
    }
    for (int t = 0; t < numTiles; ++t) {
        if (t + 1 < numTiles) {
            int an = (t + 1) * NT + tid;
            int ac = (an < NA) ? an : (NA - 1);
            unsigned int voff = (unsigned int)ac * (unsigned int)(PRED_W * 4);
            unsigned int lds  = (unsigned int)(uintptr_t)&stage[(t + 1) & 1][tid];
            __builtin_prefetch(basep + (size_t)(ac + 2 * NT) * PRED_W, 0, 1);
            asm volatile("global_load_async_to_lds_b32 %0, %1, %2\n\t"
                         "s_wait_asynccnt 1"
                         :: "v"(lds), "v"(voff), "s"(basep) : "memory");
        } else {
            asm volatile("s_wait_asynccnt 0" ::: "memory");
        }
        float x = stage[t & 1][tid];
        int a = t * NT + tid;
        if (a < NA) atomicAdd(&hist[logit_bin(x)], 1);
    }
    __syncthreads();

    int thrBin = find_thr_bin(hist, strip, tid, TOPK);

    for (int i = tid; i < CAP; i += NT) { candL[i] = -1e30f; candI[i] = -1; }
    __syncthreads();

    // ---- Pass 2: collect candidates >= threshold bin (L2-resident reread) ----
    for (int a = tid; a < NA; a += NT) {
        float x = basep[(size_t)a * PRED_W];
        if (logit_bin(x) >= thrBin) {
            int pos = atomicAdd(&cnt, 1);
            if (pos < CAP) { candL[pos] = x; candI[pos] = a; }
        }
    }
    __syncthreads();

    bitonic_desc(candL, candI, tid);

    // ---- Decode the top-100: sigmoid score + box decode against anchors ----
    if (tid < TOPK) {
        int a = candI[tid];
        float score = -1.0f;                    // padding: never kept
        float x1 = 0.f, y1 = 0.f, x2 = 0.f, y2 = 0.f;
        if (a >= 0) {
            score = sigmoidf_(candL[tid]);
            const float* bp = preds + ((size_t)b * NA + a) * PRED_W; // 16B aligned
            float4 bb = *(const float4*)bp;
            float acx, acy, aaw, aah;
            anchor_at(a, acx, acy, aaw, aah);
            float tx = bb.x * 0.1f, ty = bb.y * 0.1f;
            float tw = bb.z * 0.2f, th = bb.w * 0.2f;
            float cx = tx * aaw + acx;
            float cy = ty * aah + acy;
            float w  = __expf(tw) * aaw;
            float h  = __expf(th) * aah;
            x1 = cx - 0.5f * w; y1 = cy - 0.5f * h;
            x2 = cx + 0.5f * w; y2 = cy + 0.5f * h;
        }
        sc[tid] = score;
        bxs[0][tid] = x1; bxs[1][tid] = y1; bxs[2][tid] = x2; bxs[3][tid] = y2;
        iouMask[tid][0] = iouMask[tid][1] = iouMask[tid][2] = iouMask[tid][3] = 0u;
    }
    __syncthreads();

    // ---- Pairwise IoU > 0.5 bitmask (only j < i needed) ----
    for (int p = tid; p < TOPK * TOPK; p += NT) {
        int i = p / TOPK, j = p % TOPK;
        if (j < i) {
            float xx1 = fmaxf(bxs[0][i], bxs[0][j]);
            float yy1 = fmaxf(bxs[1][i], bxs[1][j]);
            float xx2 = fminf(bxs[2][i], bxs[2][j]);
            float yy2 = fminf(bxs[3][i], bxs[3][j]);
            float iw = fmaxf(xx2 - xx1, 0.f);
            float ih = fmaxf(yy2 - yy1, 0.f);
            float inter = iw * ih;
            float ai = (bxs[2][i] - bxs[0][i]) * (bxs[3][i] - bxs[1][i]);
            float aj = (bxs[2][j] - bxs[0][j]) * (bxs[3][j] - bxs[1][j]);
            float iou = inter / (ai + aj - inter + 1e-8f);
            if (iou > IOU_THRF)
                atomicOr(&iouMask[i][j >> 5], 1u << (j & 31));
        }
    }
    __syncthreads();

    // ---- Exact greedy NMS recurrence on one lane via bitmasks ----
    if (tid == 0) {
        unsigned int kw0 = 0, kw1 = 0, kw2 = 0, kw3 = 0;
        for (int i = 0; i < TOPK; ++i) {
            unsigned int ov = (iouMask[i][0] & kw0) | (iouMask[i][1] & kw1) |
                              (iouMask[i][2] & kw2) | (iouMask[i][3] & kw3);
            if ((sc[i] > CONF_THRF) && (ov == 0u)) {
                unsigned int bit = 1u << (i & 31);
                if      (i < 32) kw0 |= bit;
                else if (i < 64) kw1 |= bit;
                else if (i < 96) kw2 |= bit;
                else             kw3 |= bit;
            }
        }
        keepW[0] = kw0; keepW[1] = kw1; keepW[2] = kw2; keepW[3] = kw3;
    }
    __syncthreads();

    if (tid < TOPK) {
        bool kept = (keepW[tid >> 5] >> (tid & 31)) & 1u;
        size_t o = ((size_t)b * NC + c) * TOPK + tid;
        wsScores[o] = kept ? sc[tid] : -1.0f;   // reference: kept_s else -1
        float4 ob = make_float4(bxs[0][tid], bxs[1][tid], bxs[2][tid], bxs[3][tid]);
        *(float4*)(wsBoxes + o * 4) = ob;       // boxes stored regardless of keep
    }
}

// ---------------- Kernel B: per-image final top-100 + output ----------------
__global__ void __launch_bounds__(NT)
final_topk(const float* __restrict__ wsScores,
           const float* __restrict__ wsBoxes,
           float* __restrict__ out) {
    const int b   = blockIdx.x;
    const int tid = threadIdx.x;
    const int N   = NC * TOPK;   // 8000

    __shared__ int   hist[NBINS];
    __shared__ int   strip[NT];
    __shared__ float candL[CAP];
    __shared__ int   candI[CAP];
    __shared__ int   cnt;
    __shared__ int   vcount;

    for (int i = tid; i < NBINS; i += NT) hist[i] = 0;
    if (tid == 0) { cnt = 0; vcount = 0; }
    __syncthreads();

    const float* s = wsScores + (size_t)b * N;
    for (int i = tid; i < N; i += NT) {
        float v = s[i];
        if (v > 0.0f) atomicAdd(&hist[score_bin(v)], 1);
    }
    __syncthreads();

    int thrBin = find_thr_bin(hist, strip, tid, TOPK);

    for (int i = tid; i < CAP; i += NT) { candL[i] = -1e30f; candI[i] = -1; }
    __syncthreads();

    for (int i = tid; i < N; i += NT) {
        float v = s[i];
        if (v > 0.0f && score_bin(v) >= thrBin) {
            int pos = atomicAdd(&cnt, 1);
            if (pos < CAP) { candL[pos] = v; candI[pos] = i; }
        }
    }
    __syncthreads();

    bitonic_desc(candL, candI, tid);

    // Output layout (flat, in reference return order):
    //   boxes  [NB,TOPK,4] @ 0        (3200 floats)
    //   scores [NB,TOPK]   @ 3200     ( 800 floats)
    //   classes[NB,TOPK]   @ 4000     ( 800 floats)
    //   valid  [NB] int32  @ 4800     (   8 ints)
    float* oBoxes  = out;
    float* oScores = out + NB * TOPK * 4;
    float* oCls    = oScores + NB * TOPK;
    int*   oValid  = (int*)(out + NB * TOPK * 4 + NB * TOPK * 2);

    if (tid < TOPK) {
        float v   = candL[tid];
        int   idx = candI[tid];
        bool valid = (v > 0.0f) && (idx >= 0);
        float bx = 0.f, by = 0.f, bz = 0.f, bw = 0.f, cls = 0.f, scv = 0.f;
        if (valid) {
            const float* bp = wsBoxes + ((size_t)b * N + idx) * 4;
            bx = bp[0]; by = bp[1]; bz = bp[2]; bw = bp[3];
            cls = (float)(idx / TOPK);
            scv = v;
            atomicAdd(&vcount, 1);
        }
        size_t k = (size_t)b * TOPK + tid;
        oBoxes[k * 4 + 0] = bx;
        oBoxes[k * 4 + 1] = by;
        oBoxes[k * 4 + 2] = bz;
        oBoxes[k * 4 + 3] = bw;
        oScores[k] = scv;
        oCls[k]    = cls;
    }
    __syncthreads();
    if (tid == 0) oValid[b] = vcount;
}

// ---------------- host entry ----------------
extern "C" void kernel_launch(void* const* d_in, const int* in_sizes, int n_in,
                              void* d_out, int out_size, void* d_ws, size_t ws_size,
                              hipStream_t stream) {
    (void)in_sizes; (void)n_in; (void)out_size; (void)ws_size;
    // inputs: d_in[0]=images (unused by math), d_in[1]=predictions [8,76725,84] f32
    const float* preds = (const float*)d_in[1];

    float* ws       = (float*)d_ws;
    float* wsScores = ws;                                   //  64,000 floats
    float* wsBoxes  = ws + (size_t)NB * NC * TOPK;          // 256,000 floats

    dim3 gA(NC, NB);
    topk_nms_per_class<<<gA, NT, 0, stream>>>(preds, wsScores, wsBoxes);
    final_topk<<<NB, NT, 0, stream>>>(wsScores, wsBoxes, (float*)d_out);
}